// TWIRLS_19696720019620
// MI455X (gfx1250) — compile-verified
//
#include <hip/hip_runtime.h>
#include <hip/hip_bf16.h>

typedef __attribute__((ext_vector_type(2))) float v2f;
typedef __attribute__((ext_vector_type(8))) float v8f;

#define IN_SIZE   256
#define HIDDEN    128
#define OUT_SIZE  40
#define NUM_STEPS 16
#define LAM_F     1.0f
#define ALPHA_F   0.5f

// ---------------------------------------------------------------- utilities
__global__ void twirls_zero_f32(float* __restrict__ p, int n) {
    int i = blockIdx.x * blockDim.x + threadIdx.x;
    if (i < n) p[i] = 0.0f;
}

__global__ void twirls_degree(const int* __restrict__ row, float* __restrict__ deg, int E) {
    int e = blockIdx.x * blockDim.x + threadIdx.x;
    if (e < E) atomicAdd(&deg[row[e]], 1.0f);   // exact integer-valued adds
}

__global__ void twirls_invd(float* __restrict__ deg, int N) {
    int i = blockIdx.x * blockDim.x + threadIdx.x;
    if (i < N) deg[i] = 1.0f / (LAM_F * deg[i] + 1.0f);
}

// ------------------------------------------------- fp32 WMMA GEMM (16x16x4)
// C[M,NCOLS] = act(A[M,K] @ B[K,NCOLS] + bias). One wave per 16x16 tile.
// M must be a multiple of 16 (50000 = 3125*16). K, NCOLS compile-time.
// NCOLS%16 != 0 handled branchlessly: clamp column + 0/1 mask (no exec
// manipulation in the K-loop -> no saveexec/branch per load).
template<int K, int NCOLS, bool RELU>
__global__ void twirls_gemm_wmma(const float* __restrict__ A,
                                 const float* __restrict__ B,
                                 const float* __restrict__ bias,
                                 float* __restrict__ C,
                                 int nTilesN, int nTilesTotal) {
    constexpr bool GUARD = (NCOLS % 16) != 0;

    const int lane = threadIdx.x & 31;
    const int wave = threadIdx.x >> 5;
    const int tile = blockIdx.x * (blockDim.x >> 5) + wave;
    if (tile >= nTilesTotal) return;          // whole-wave exit; EXEC all-1s after
    const int tileM = tile / nTilesN;
    const int tileN = tile - tileM * nTilesN;

    const int mloc = lane & 15;               // row within A tile / col within B,C tile
    const int hi   = lane >> 4;               // half-wave selector (K pair / M half)
    const int n    = tileN * 16 + mloc;       // output column

    int   nc  = n;                            // clamped, always-in-range column
    float msk = 1.0f;
    if (GUARD) {
        const bool nok = (n < NCOLS);
        nc  = nok ? n : (NCOLS - 1);          // v_cndmask, no branch
        msk = nok ? 1.0f : 0.0f;
    }

    const float* __restrict__ arow = A + (size_t)(tileM * 16 + mloc) * (size_t)K;

    v8f c0 = {0.f, 0.f, 0.f, 0.f, 0.f, 0.f, 0.f, 0.f};
    v8f c1 = {0.f, 0.f, 0.f, 0.f, 0.f, 0.f, 0.f, 0.f};

    #pragma unroll 2
    for (int k = 0; k < K; k += 8) {
        {   // chunk k..k+3 -> c0
            const int ka = k + 2 * hi;
            v2f a, b;
            a.x = arow[ka];
            a.y = arow[ka + 1];
            b.x = B[(size_t)ka       * NCOLS + nc];
            b.y = B[(size_t)(ka + 1) * NCOLS + nc];
            if (GUARD) { b.x *= msk; b.y *= msk; }
            c0 = __builtin_amdgcn_wmma_f32_16x16x4_f32(
                     false, a, false, b, (short)0, c0, false, false);
        }
        {   // chunk k+4..k+7 -> c1 (independent accumulator, no RAW stall)
            const int ka = k + 4 + 2 * hi;
            v2f a, b;
            a.x = arow[ka];
            a.y = arow[ka + 1];
            b.x = B[(size_t)ka       * NCOLS + nc];
            b.y = B[(size_t)(ka + 1) * NCOLS + nc];
            if (GUARD) { b.x *= msk; b.y *= msk; }
            c1 = __builtin_amdgcn_wmma_f32_16x16x4_f32(
                     false, a, false, b, (short)0, c1, false, false);
        }
    }

    if (!GUARD || n < NCOLS) {                // single guard region, outside hot loop
        const float bv = bias[n];
        #pragma unroll
        for (int r = 0; r < 8; ++r) {
            float v = c0[r] + c1[r] + bv;     // C layout: VGPR r -> M = r + 8*hi
            if (RELU) v = fmaxf(v, 0.f);
            C[(size_t)(tileM * 16 + r + 8 * hi) * NCOLS + n] = v;
        }
    }
}

// -------------------------------------------------------- edge scatter SpMM
// AY[row[e]][:] += Y[col[e]][:]; 32 lanes per edge, float4 per lane.
__global__ void twirls_spmm(const int* __restrict__ row, const int* __restrict__ col,
                            const float* __restrict__ Y, float* __restrict__ AY, int E) {
    int tid = blockIdx.x * blockDim.x + threadIdx.x;
    int e   = tid >> 5;
    if (e >= E) return;
    int j = (tid & 31) * 4;
    float4 v = *(const float4*)(Y + (size_t)col[e] * HIDDEN + j);
    float* dst = AY + (size_t)row[e] * HIDDEN + j;
    atomicAdd(dst + 0, v.x);
    atomicAdd(dst + 1, v.y);
    atomicAdd(dst + 2, v.z);
    atomicAdd(dst + 3, v.w);
}

// ---------------------------------------------- fused blend + AY reset to 0
__global__ void twirls_update(float* __restrict__ Y, float* __restrict__ AY,
                              const float* __restrict__ Y0,
                              const float* __restrict__ inv_d, int N) {
    int tid = blockIdx.x * blockDim.x + threadIdx.x;
    int i   = tid >> 5;
    if (i >= N) return;
    int j = (tid & 31) * 4;
    size_t off = (size_t)i * HIDDEN + j;
    float  id  = inv_d[i];
    float4 y   = *(float4*)(Y  + off);
    float4 ay  = *(float4*)(AY + off);
    float4 y0  = *(const float4*)(Y0 + off);
    float4 o;
    o.x = (1.f - ALPHA_F) * y.x + ALPHA_F * id * (LAM_F * ay.x + y0.x);
    o.y = (1.f - ALPHA_F) * y.y + ALPHA_F * id * (LAM_F * ay.y + y0.y);
    o.z = (1.f - ALPHA_F) * y.z + ALPHA_F * id * (LAM_F * ay.z + y0.z);
    o.w = (1.f - ALPHA_F) * y.w + ALPHA_F * id * (LAM_F * ay.w + y0.w);
    *(float4*)(Y + off) = o;
    float4 z = {0.f, 0.f, 0.f, 0.f};
    *(float4*)(AY + off) = z;                 // ready for next step's scatter
}

// --------------------------------------------------------------------------
extern "C" void kernel_launch(void* const* d_in, const int* in_sizes, int n_in,
                              void* d_out, int out_size, void* d_ws, size_t ws_size,
                              hipStream_t stream) {
    const float* X    = (const float*)d_in[0];
    const int*   row  = (const int*)  d_in[1];
    const int*   col  = (const int*)  d_in[2];
    const float* W1   = (const float*)d_in[3];
    const float* b1   = (const float*)d_in[4];
    const float* W2   = (const float*)d_in[5];
    const float* b2   = (const float*)d_in[6];
    const float* Wout = (const float*)d_in[7];
    const float* bout = (const float*)d_in[8];
    float*       out  = (float*)d_out;

    const int N = in_sizes[0] / IN_SIZE;      // 50000 (multiple of 16)
    const int E = in_sizes[1];                // 800000

    // workspace layout (floats): inv_d | Y0 | Y | H-then-AY   (~77 MB total)
    float* ws    = (float*)d_ws;
    size_t nPad  = ((size_t)N + 255) & ~(size_t)255;
    float* inv_d = ws;
    float* Y0    = ws + nPad;
    float* Y     = Y0 + (size_t)N * HIDDEN;
    float* HA    = Y  + (size_t)N * HIDDEN;   // H for the MLP, then AY scratch

    const int nTilesM  = N / 16;              // 3125
    const int tnHid    = HIDDEN / 16;         // 8
    const int tnOut    = (OUT_SIZE + 15) / 16;// 3
    const int tilesHid = nTilesM * tnHid;     // 25000
    const int tilesOut = nTilesM * tnOut;     // 9375
    const int WPB      = 256 / 32;            // 8 waves per block

    // degree -> inv_d
    twirls_zero_f32<<<(N + 255) / 256, 256, 0, stream>>>(inv_d, N);
    twirls_degree  <<<(E + 255) / 256, 256, 0, stream>>>(row, inv_d, E);
    twirls_invd    <<<(N + 255) / 256, 256, 0, stream>>>(inv_d, N);

    // MLP: H = relu(X@W1 + b1);  Y0 = H@W2 + b2
    twirls_gemm_wmma<IN_SIZE, HIDDEN, true ><<<(tilesHid + WPB - 1) / WPB, 256, 0, stream>>>(
        X,  W1, b1, HA, tnHid, tilesHid);
    twirls_gemm_wmma<HIDDEN,  HIDDEN, false><<<(tilesHid + WPB - 1) / WPB, 256, 0, stream>>>(
        HA, W2, b2, Y0, tnHid, tilesHid);

    // Y = Y0 ; AY = 0
    hipMemcpyAsync(Y, Y0, (size_t)N * HIDDEN * sizeof(float),
                   hipMemcpyDeviceToDevice, stream);
    twirls_zero_f32<<<((N * HIDDEN) + 255) / 256, 256, 0, stream>>>(HA, N * HIDDEN);

    // 16 propagation steps (L2-resident working set)
    for (int s = 0; s < NUM_STEPS; ++s) {
        twirls_spmm  <<<((E * 32) + 255) / 256, 256, 0, stream>>>(row, col, Y, HA, E);
        twirls_update<<<((N * 32) + 255) / 256, 256, 0, stream>>>(Y, HA, Y0, inv_d, N);
    }

    // out = Y @ Wout + bout
    twirls_gemm_wmma<HIDDEN, OUT_SIZE, false><<<(tilesOut + WPB - 1) / WPB, 256, 0, stream>>>(
        Y, Wout, bout, out, tnOut, tilesOut);
}